// DGCNN_MaskFeat_36764920054486
// MI455X (gfx1250) — compile-verified
//
#include <hip/hip_runtime.h>
#include <hip/hip_bf16.h>

typedef __attribute__((ext_vector_type(2))) float v2f;
typedef __attribute__((ext_vector_type(8))) float v8f;

#define BATCH 8
#define NPTS  2048
#define KNN_K 20
#define NEG_SLOPE 0.2f

// ---------------------------------------------------------------------------
// kNN: one thread per query point, queries staged in LDS (padded stride),
// candidate rows are wave-uniform broadcast loads. Per-thread top-20 list.
// ---------------------------------------------------------------------------
template <int C>
__global__ void __launch_bounds__(64)
knn_kernel(const float* __restrict__ x, int* __restrict__ idx_out) {
    const int b  = blockIdx.x / (NPTS / 64);
    const int n0 = (blockIdx.x % (NPTS / 64)) * 64;
    const int t  = threadIdx.x;
    constexpr int CP = C + 1;                 // pad stride -> conflict-free
    __shared__ float q[64 * CP];

    const float* xb = x + (size_t)b * NPTS * C;
    for (int e = t; e < 64 * C; e += 64) {
        int r = e / C, c = e % C;
        q[r * CP + c] = xb[(size_t)(n0 + r) * C + c];
    }
    __syncthreads();

    float bd[KNN_K];
    int   bi[KNN_K];
#pragma unroll
    for (int j = 0; j < KNN_K; ++j) { bd[j] = 3.4e38f; bi[j] = 0; }

    const float* qp = q + t * CP;
    for (int m = 0; m < NPTS; ++m) {
        const float* xm = xb + (size_t)m * C;
        float d = 0.f;
#pragma unroll 4
        for (int c = 0; c < C; ++c) {
            float df = qp[c] - xm[c];
            d += df * df;
        }
        if (d < bd[KNN_K - 1]) {
            int j = KNN_K - 1;
            while (j > 0 && bd[j - 1] > d) {
                bd[j] = bd[j - 1];
                bi[j] = bi[j - 1];
                --j;
            }
            bd[j] = d;
            bi[j] = m;
        }
    }

    int* op = idx_out + ((size_t)b * NPTS + n0 + t) * KNN_K;
#pragma unroll
    for (int j = 0; j < KNN_K; ++j) op[j] = bi[j];
}

// ---------------------------------------------------------------------------
// EdgeConv: block = 4 wave32 = 16 points x all COUT columns.
// Per neighbor k: stage 16 edge-feature rows in LDS, WMMA f32 16x16x4 over
// the 2*CIN reduction dim, fused scale/bias/LeakyReLU + running max over k.
// ---------------------------------------------------------------------------
template <int CIN, int COUT>
__global__ void __launch_bounds__(128)
edgeconv_kernel(const float* __restrict__ xin, const int* __restrict__ idx,
                const float* __restrict__ W, const float* __restrict__ g,
                const float* __restrict__ bias, float* __restrict__ xout) {
    constexpr int FD  = 2 * CIN;              // edge feature dim
    constexpr int FDP = (FD + 3) & ~3;        // padded to K-step multiple
    constexpr int FS  = FDP + 4;              // LDS row stride (bank spread)
    constexpr int NT  = COUT / 16;            // 16-wide column tiles
    constexpr int TPW = NT / 4;               // tiles per wave (4 waves)

    const int b    = blockIdx.x / (NPTS / 16);
    const int n0   = (blockIdx.x % (NPTS / 16)) * 16;
    const int t    = threadIdx.x;
    const int wave = t >> 5;
    const int lane = t & 31;
    const int M    = lane & 15;               // A row / B,D column within tile
    const int hi   = lane >> 4;               // lane half selects K pair / D rows

    __shared__ float feat[16 * FS];
    const float* xb   = xin + (size_t)b * NPTS * CIN;
    const int*   idxb = idx + ((size_t)b * NPTS + n0) * KNN_K;

    v8f   maxacc[TPW];
    float gv[TPW], bv[TPW];
#pragma unroll
    for (int tt = 0; tt < TPW; ++tt) {
        int col = (wave * TPW + tt) * 16 + M;
        gv[tt] = g[col];
        bv[tt] = bias[col];
#pragma unroll
        for (int v = 0; v < 8; ++v) maxacc[tt][v] = -3.4e38f;
    }

    for (int k = 0; k < KNN_K; ++k) {
        __syncthreads();                      // previous tile fully consumed
        for (int e = t; e < 16 * FDP; e += 128) {
            int r = e / FDP, c = e % FDP;
            float v = 0.f;
            if (c < FD) {
                int n = n0 + r;
                if (c < CIN) {
                    int nb = idxb[r * KNN_K + k];
                    v = xb[(size_t)nb * CIN + c] - xb[(size_t)n * CIN + c];
                } else {
                    v = xb[(size_t)n * CIN + (c - CIN)];
                }
            }
            feat[r * FS + c] = v;
        }
        __syncthreads();

#pragma unroll
        for (int tt = 0; tt < TPW; ++tt) {
            const int col0 = (wave * TPW + tt) * 16;
            v8f acc = {};
#pragma unroll
            for (int kc = 0; kc < FDP / 4; ++kc) {
                const int kb = kc * 4 + hi * 2;
                v2f a, bb;
                a.x  = feat[M * FS + kb];
                a.y  = feat[M * FS + kb + 1];
                bb.x = (kb     < FD) ? W[(size_t)kb * COUT + col0 + M]       : 0.f;
                bb.y = (kb + 1 < FD) ? W[(size_t)(kb + 1) * COUT + col0 + M] : 0.f;
                acc = __builtin_amdgcn_wmma_f32_16x16x4_f32(
                        false, a, false, bb, (short)0, acc, false, false);
            }
#pragma unroll
            for (int v = 0; v < 8; ++v) {
                float h = acc[v] * gv[tt] + bv[tt];
                h = (h >= 0.f) ? h : NEG_SLOPE * h;
                maxacc[tt][v] = fmaxf(maxacc[tt][v], h);
            }
        }
    }

    float* ob = xout + ((size_t)b * NPTS + n0) * COUT;
#pragma unroll
    for (int tt = 0; tt < TPW; ++tt) {
        int col = (wave * TPW + tt) * 16 + M;
#pragma unroll
        for (int v = 0; v < 8; ++v) {
            int row = hi * 8 + v;
            ob[(size_t)row * COUT + col] = maxacc[tt][v];
        }
    }
}

// ---------------------------------------------------------------------------
// Final fused layer: concat[x1|x2|x3|x4] (512) x W5 (512x1024), scale/bias/
// LeakyReLU, transposed store to (B, 1024, N). Concat staged in LDS directly.
// ---------------------------------------------------------------------------
__global__ void __launch_bounds__(128)
final_kernel(const float* __restrict__ x1, const float* __restrict__ x2,
             const float* __restrict__ x3, const float* __restrict__ x4,
             const float* __restrict__ W, const float* __restrict__ g,
             const float* __restrict__ bias, float* __restrict__ out) {
    constexpr int CF = 512, CO = 1024, FS = CF + 4;
    const int b    = blockIdx.x / (NPTS / 16);
    const int n0   = (blockIdx.x % (NPTS / 16)) * 16;
    const int t    = threadIdx.x;
    const int wave = t >> 5;
    const int lane = t & 31;
    const int M    = lane & 15;
    const int hi   = lane >> 4;

    __shared__ float feat[16 * FS];
    for (int e = t; e < 16 * CF; e += 128) {
        int r = e / CF, c = e % CF;
        size_t n = (size_t)b * NPTS + n0 + r;
        float v;
        if (c < 64)       v = x1[n * 64  + c];
        else if (c < 128) v = x2[n * 64  + (c - 64)];
        else if (c < 256) v = x3[n * 128 + (c - 128)];
        else              v = x4[n * 256 + (c - 256)];
        feat[r * FS + c] = v;
    }
    __syncthreads();

    const int col0 = blockIdx.y * 64 + wave * 16;
    v8f acc = {};
#pragma unroll 8
    for (int kc = 0; kc < CF / 4; ++kc) {
        const int kb = kc * 4 + hi * 2;
        v2f a, bb;
        a.x  = feat[M * FS + kb];
        a.y  = feat[M * FS + kb + 1];
        bb.x = W[(size_t)kb * CO + col0 + M];
        bb.y = W[(size_t)(kb + 1) * CO + col0 + M];
        acc = __builtin_amdgcn_wmma_f32_16x16x4_f32(
                false, a, false, bb, (short)0, acc, false, false);
    }

    const int col = col0 + M;
    const float gvv = g[col], bvv = bias[col];
    float* ob = out + (size_t)b * CO * NPTS;       // out[b][c][n]
#pragma unroll
    for (int v = 0; v < 8; ++v) {
        float h = acc[v] * gvv + bvv;
        h = (h >= 0.f) ? h : NEG_SLOPE * h;
        int row = hi * 8 + v;
        ob[(size_t)col * NPTS + (n0 + row)] = h;
    }
}

// ---------------------------------------------------------------------------
extern "C" void kernel_launch(void* const* d_in, const int* in_sizes, int n_in,
                              void* d_out, int out_size, void* d_ws, size_t ws_size,
                              hipStream_t stream) {
    const float* pts = (const float*)d_in[0];
    const float* W1 = (const float*)d_in[1];
    const float* g1 = (const float*)d_in[2];
    const float* b1 = (const float*)d_in[3];
    const float* W2 = (const float*)d_in[4];
    const float* g2 = (const float*)d_in[5];
    const float* b2 = (const float*)d_in[6];
    const float* W3 = (const float*)d_in[7];
    const float* g3 = (const float*)d_in[8];
    const float* b3 = (const float*)d_in[9];
    const float* W4 = (const float*)d_in[10];
    const float* g4 = (const float*)d_in[11];
    const float* b4 = (const float*)d_in[12];
    const float* W5 = (const float*)d_in[13];
    const float* g5 = (const float*)d_in[14];
    const float* b5 = (const float*)d_in[15];
    float* out = (float*)d_out;

    // workspace carve: idx + x1..x4  (~34 MB total, L2-resident)
    char* p = (char*)d_ws;
    int*   idx = (int*)p;   p += (size_t)BATCH * NPTS * KNN_K * sizeof(int);
    float* x1  = (float*)p; p += (size_t)BATCH * NPTS * 64  * sizeof(float);
    float* x2  = (float*)p; p += (size_t)BATCH * NPTS * 64  * sizeof(float);
    float* x3  = (float*)p; p += (size_t)BATCH * NPTS * 128 * sizeof(float);
    float* x4  = (float*)p; p += (size_t)BATCH * NPTS * 256 * sizeof(float);

    const dim3 knnGrid(BATCH * NPTS / 64);
    const dim3 ecGrid(BATCH * NPTS / 16);

    knn_kernel<3><<<knnGrid, 64, 0, stream>>>(pts, idx);
    edgeconv_kernel<3, 64><<<ecGrid, 128, 0, stream>>>(pts, idx, W1, g1, b1, x1);

    knn_kernel<64><<<knnGrid, 64, 0, stream>>>(x1, idx);
    edgeconv_kernel<64, 64><<<ecGrid, 128, 0, stream>>>(x1, idx, W2, g2, b2, x2);

    knn_kernel<64><<<knnGrid, 64, 0, stream>>>(x2, idx);
    edgeconv_kernel<64, 128><<<ecGrid, 128, 0, stream>>>(x2, idx, W3, g3, b3, x3);

    knn_kernel<128><<<knnGrid, 64, 0, stream>>>(x3, idx);
    edgeconv_kernel<128, 256><<<ecGrid, 128, 0, stream>>>(x3, idx, W4, g4, b4, x4);

    final_kernel<<<dim3(BATCH * NPTS / 16, 16), 128, 0, stream>>>(
        x1, x2, x3, x4, W5, g5, b5, out);
}